// GCN_67826123538777
// MI455X (gfx1250) — compile-verified
//
#include <hip/hip_runtime.h>
#include <hip/hip_bf16.h>

// GCN on MI455X (gfx1250), compile-only target.
//
// Per layer:  Y = H @ W    -> blocked fp32 WMMA GEMM (V_WMMA_F32_16X16X4_F32),
//                             A and B tiles staged in LDS, double-buffered,
//                             register-prefetched: hot loop = LDS + WMMA only
//             H' = relu?( dinv[i] * sum_{j in N(i)} dinv[j]*Y[j,:] + b )
// Adjacency kept as an 8192x8192 bitmap (8 MB) -> set-semantics dedup for free,
// degrees via popcount. Whole working set (~40 MB) is L2-resident (192 MB L2).

typedef __attribute__((ext_vector_type(2))) float v2f;
typedef __attribute__((ext_vector_type(8))) float v8f;

#define GCN_N 8192
#define ROW_WORDS 256   // 8192 bits / 32

#define KT      32      // K-depth of one staged tile
#define BROW    136     // B LDS row stride (words): 2-row half-offset = 272%64=16
                        //  -> lane halves hit disjoint 16-bank spans
#define ASTRIDE 36      // A LDS row stride (words): m*36%64 covers 16 distinct
                        //  multiples of 4; +2h fills the pair gaps -> b64 reads
                        //  from 32 lanes tile all 64 banks exactly once

// ---------------------------------------------------------------------------
// 1) init bitmap (zeros + self-loop diagonal bits)
// ---------------------------------------------------------------------------
__global__ void gcn_init_bitmap(unsigned* __restrict__ bm, int total_words) {
  int idx = blockIdx.x * blockDim.x + threadIdx.x;
  if (idx >= total_words) return;
  int row = idx >> 8;          // /ROW_WORDS
  int w   = idx & (ROW_WORDS - 1);
  unsigned v = ((row >> 5) == w) ? (1u << (row & 31)) : 0u;  // self loop
  bm[idx] = v;
}

// ---------------------------------------------------------------------------
// 2) scatter edges (symmetric) into bitmap; atomicOr collapses duplicates
// ---------------------------------------------------------------------------
__global__ void gcn_build_edges(const int* __restrict__ e0,
                                const int* __restrict__ e1,
                                unsigned* __restrict__ bm, int nedges) {
  int e = blockIdx.x * blockDim.x + threadIdx.x;
  if (e >= nedges) return;
  int a = e0[e];
  int b = e1[e];
  atomicOr(&bm[(size_t)a * ROW_WORDS + (b >> 5)], 1u << (b & 31));
  atomicOr(&bm[(size_t)b * ROW_WORDS + (a >> 5)], 1u << (a & 31));
}

// ---------------------------------------------------------------------------
// 3) degrees -> dinv = 1/sqrt(deg).  One wave32 per row, popcount + shfl_xor.
// ---------------------------------------------------------------------------
__global__ void gcn_degrees(const unsigned* __restrict__ bm,
                            float* __restrict__ dinv, int nrows) {
  int wave = (blockIdx.x * blockDim.x + threadIdx.x) >> 5;
  int lane = threadIdx.x & 31;
  if (wave >= nrows) return;
  const unsigned* row = bm + (size_t)wave * ROW_WORDS;
  int cnt = 0;
#pragma unroll
  for (int w = 0; w < ROW_WORDS; w += 32) cnt += __popc(row[w + lane]);
#pragma unroll
  for (int off = 16; off > 0; off >>= 1) cnt += __shfl_xor(cnt, off, 32);
  if (lane == 0) dinv[wave] = rsqrtf((float)cnt);
}

// ---------------------------------------------------------------------------
// 4) GEMM  C[M x Nc] = A[M x K] * B[K x Nc]   (fp32, WMMA 16x16x4)
//
//    Block = 256 threads = 8 waves; block tile 64(M) x 128(N).
//    Wave w: wm = w>>1 (16-row band), wn = w&1 (64-col band); each wave owns a
//    16x64 C tile = 4 x v8f accumulators.
//    A tile (64 x KT) and B tile (KT x 128) both staged in LDS, double
//    buffered; next tiles are prefetched into registers before the 32 WMMAs of
//    the current tile and stored to the alternate buffers after, so global
//    latency fully overlaps matrix work. Steady-state loop is ds_load + wmma.
//
//    Fragment layouts (lane L, m=L&15, h=L>>4):
//      A 16x4 f32 (2 VGPRs): a = { A[m][k+2h], A[m][k+2h+1] }  (v2f ds_load_b64)
//      B 4x16     (2 VGPRs): b = { B[k+2h][n], B[k+2h+1][n] }  (ds_load_2addr)
//      C 16x16    (8 VGPRs): VGPR r -> row r+8h, col m
//    All tile counts divide exactly -> no partial waves -> EXEC all-1s.
// ---------------------------------------------------------------------------
__global__ __launch_bounds__(256) void gcn_gemm_wmma_f32(
    const float* __restrict__ A, const float* __restrict__ B,
    float* __restrict__ C, int M, int K, int Nc) {
  __shared__ float bsB[2][KT * BROW];    // 2 x 17 KB
  __shared__ float bsA[2][64 * ASTRIDE]; // 2 x  9 KB

  const int tid  = threadIdx.x;
  const int lane = tid & 31;
  const int wave = tid >> 5;
  const int wm   = wave >> 1;          // 0..3
  const int wn   = wave & 1;           // 0..1

  const int tilesN = Nc >> 7;          // 128-wide block tiles
  const int bm0 = (blockIdx.x / tilesN) * 64;
  const int bn0 = (blockIdx.x % tilesN) * 128;

  const int m = lane & 15;
  const int h = lane >> 4;

  // cooperative B fill: 8 threads per row (KT rows), 16 consecutive cols each
  const int blr = tid >> 3;            // 0..31 : row within KT tile
  const int blc = (tid & 7) * 16;      // col within 128
  // cooperative A fill: 4 threads per row (64 rows), 8 consecutive k each
  const int alr = tid >> 2;            // 0..63 : row within 64-row band
  const int alc = (tid & 3) * 8;       // k-offset within KT

  const float* aptr = A + (size_t)(bm0 + alr) * K + alc;
  const float* bptr = B + (size_t)blr * Nc + bn0 + blc;

  // prologue: tile 0 -> LDS buffer 0
  {
    float4 rb[4], ra[2];
#pragma unroll
    for (int j = 0; j < 4; ++j) rb[j] = *(const float4*)(bptr + 4 * j);
#pragma unroll
    for (int j = 0; j < 2; ++j) ra[j] = *(const float4*)(aptr + 4 * j);
#pragma unroll
    for (int j = 0; j < 4; ++j)
      *(float4*)&bsB[0][blr * BROW + blc + 4 * j] = rb[j];
#pragma unroll
    for (int j = 0; j < 2; ++j)
      *(float4*)&bsA[0][alr * ASTRIDE + alc + 4 * j] = ra[j];
  }
  __syncthreads();

  v8f acc[4] = {};
  const int nt = K / KT;
  for (int kt = 0; kt < nt; ++kt) {
    // prefetch next tiles into registers (overlaps the WMMAs below)
    float4 nb[4], na[2];
    if (kt + 1 < nt) {
      const float* pb = bptr + (size_t)(kt + 1) * KT * Nc;
      const float* pa = aptr + (kt + 1) * KT;
#pragma unroll
      for (int j = 0; j < 4; ++j) nb[j] = *(const float4*)(pb + 4 * j);
#pragma unroll
      for (int j = 0; j < 2; ++j) na[j] = *(const float4*)(pa + 4 * j);
    }

    const float* bb = &bsB[kt & 1][0];
    const float* ba = &bsA[kt & 1][(size_t)(wm * 16 + m) * ASTRIDE + 2 * h];
#pragma unroll
    for (int s = 0; s < KT / 4; ++s) {
      v2f a = *(const v2f*)(ba + 4 * s);           // ds_load_b64
      const int kk = 4 * s + 2 * h;
#pragma unroll
      for (int i = 0; i < 4; ++i) {
        v2f b;
        b.x = bb[(size_t)kk * BROW + wn * 64 + i * 16 + m];
        b.y = bb[(size_t)(kk + 1) * BROW + wn * 64 + i * 16 + m];
        acc[i] = __builtin_amdgcn_wmma_f32_16x16x4_f32(
            false, a, false, b, (short)0, acc[i], false, false);
      }
    }

    // publish next tiles into the other buffers
    if (kt + 1 < nt) {
#pragma unroll
      for (int j = 0; j < 4; ++j)
        *(float4*)&bsB[(kt + 1) & 1][blr * BROW + blc + 4 * j] = nb[j];
#pragma unroll
      for (int j = 0; j < 2; ++j)
        *(float4*)&bsA[(kt + 1) & 1][alr * ASTRIDE + alc + 4 * j] = na[j];
    }
    __syncthreads();
  }

  // store C
#pragma unroll
  for (int i = 0; i < 4; ++i) {
#pragma unroll
    for (int r = 0; r < 8; ++r) {
      C[(size_t)(bm0 + wm * 16 + r + 8 * h) * Nc + bn0 + wn * 64 + i * 16 + m] =
          acc[i][r];
    }
  }
}

// ---------------------------------------------------------------------------
// 5) sparse aggregation:  out[i,t] = act( dinv[i]*sum_j dinv[j]*Y[j,t] + b[t] )
//    One block per row, one thread per feature column (F = 512 or 256).
//    Bitmap words are block-uniform (broadcast); Y gathers are coalesced.
// ---------------------------------------------------------------------------
template <bool RELU>
__global__ void gcn_aggregate(const unsigned* __restrict__ bm,
                              const float* __restrict__ dinv,
                              const float* __restrict__ Y,
                              const float* __restrict__ bias,
                              float* __restrict__ out, int F) {
  int row = blockIdx.x;
  int t   = threadIdx.x;
  const unsigned* r = bm + (size_t)row * ROW_WORDS;
  float acc = 0.f;
  for (int w = 0; w < ROW_WORDS; ++w) {
    unsigned bits = r[w];
    while (bits) {
      int j = (w << 5) + __ffs(bits) - 1;
      bits &= bits - 1;
      acc = fmaf(dinv[j], Y[(size_t)j * F + t], acc);
    }
  }
  float v = fmaf(acc, dinv[row], bias[t]);
  if (RELU) v = fmaxf(v, 0.f);
  out[(size_t)row * F + t] = v;
}

// ---------------------------------------------------------------------------
// launch
// ---------------------------------------------------------------------------
extern "C" void kernel_launch(void* const* d_in, const int* in_sizes, int n_in,
                              void* d_out, int out_size, void* d_ws, size_t ws_size,
                              hipStream_t stream) {
  const float* x  = (const float*)d_in[0];
  const int*   ei = (const int*)d_in[1];     // (2, E) flat, int32
  const float* W1 = (const float*)d_in[2];
  const float* b1 = (const float*)d_in[3];
  const float* W2 = (const float*)d_in[4];
  const float* b2 = (const float*)d_in[5];
  const float* W3 = (const float*)d_in[6];
  const float* b3 = (const float*)d_in[7];
  float* out = (float*)d_out;

  const int N = GCN_N;
  const int F = 512;
  const int nedges = in_sizes[1] / 2;
  const int* e0 = ei;
  const int* e1 = ei + nedges;

  // workspace layout (needs ~40.03 MB):
  //   [0,  8MB)   adjacency bitmap  (N * 256 u32)
  //   [8MB, +32K) dinv              (N f32)
  //   then Y (N*512 f32, 16MB), H (N*512 f32, 16MB)
  unsigned char* ws = (unsigned char*)d_ws;
  unsigned* bm  = (unsigned*)ws;
  float* dinv   = (float*)(ws + (size_t)N * ROW_WORDS * sizeof(unsigned));
  float* Y      = (float*)((unsigned char*)dinv + (size_t)N * sizeof(float));
  float* H      = Y + (size_t)N * F;

  // --- build normalized adjacency ---
  {
    int total_words = N * ROW_WORDS;
    gcn_init_bitmap<<<(total_words + 255) / 256, 256, 0, stream>>>(bm, total_words);
    gcn_build_edges<<<(nedges + 255) / 256, 256, 0, stream>>>(e0, e1, bm, nedges);
    gcn_degrees<<<(N * 32 + 255) / 256, 256, 0, stream>>>(bm, dinv, N);
  }

  // --- layer 1: H = relu(Ahat @ (x W1) + b1) ---
  {
    int blocks = (N / 64) * (F / 128);                    // 512
    gcn_gemm_wmma_f32<<<blocks, 256, 0, stream>>>(x, W1, Y, N, 512, F);
    gcn_aggregate<true><<<N, F, 0, stream>>>(bm, dinv, Y, b1, H, F);
  }
  // --- layer 2: H = relu(Ahat @ (H W2) + b2) ---
  {
    int blocks = (N / 64) * (F / 128);
    gcn_gemm_wmma_f32<<<blocks, 256, 0, stream>>>(H, W2, Y, N, 512, F);
    gcn_aggregate<true><<<N, F, 0, stream>>>(bm, dinv, Y, b2, H, F);
  }
  // --- layer 3: out = Ahat @ (H W3) + b3   (F_out = 256) ---
  {
    const int Fo = 256;
    int blocks = (N / 64) * (Fo / 128);                   // 256
    gcn_gemm_wmma_f32<<<blocks, 256, 0, stream>>>(H, W3, Y, N, 512, Fo);
    gcn_aggregate<false><<<N, Fo, 0, stream>>>(bm, dinv, Y, b3, out, Fo);
  }
}